// TropcialBottleneck_53429393162415
// MI455X (gfx1250) — compile-verified
//
#include <hip/hip_runtime.h>
#include <cstdint>

// ---------------------------------------------------------------------------
// Tropical (min-plus / max-plus) tiled GEMM for MI455X (gfx1250).
//
//   y[b,i] = RED_j ( x[b,j] + W[i,j] )   [+ bias[i]]  [min/max with aux[b,i]]
//
// RED is min or max -> fundamentally not expressible on WMMA/SWMMAC units
// (they only do *,+ accumulate), so the inner loop is VALU v_add_f32 +
// v_min3/max3_num_f32 at fp32. Working set ~4 MB (L2-resident, 192 MB) and
// ~3.2 G VALU ops -> compute bound, fp32 is the right precision.
// CDNA5-specific leverage is data movement:
//   * global_load_async_to_lds_b32 (GVS: sgpr tile base + 32-bit lane offset)
//     with per-lane LDS destinations -> transposing tile copy for free
//     (tiles land as [BK][BM+pad] so compute reads are ds_load_b128)
//   * s_wait_asynccnt gates a 2-deep double-buffered pipeline
// Tiling: 256 threads (8 wave32), 64x64 tile, BK=32, 4x4 register micro-tile
// -> per 2 k-steps: 4x ds_load_b128 + 32 v_add + 16 v_min3/max3 per thread.
// ---------------------------------------------------------------------------

#define BM 64
#define BN 64
#define BK 32
#define LDST 68                      // BM + 4 floats pad: conflict-light, 16B aligned
#define TILEF (BK * LDST)            // 2176 floats per tile buffer
#define SMEM_BYTES (4 * TILEF * 4)   // x[2] + w[2] double buffers = 34816 B

// LDS byte bases inside the (purely dynamic => hw offset 0) LDS block.
__device__ __forceinline__ uint32_t xs_byte(int buf) { return (uint32_t)(buf * TILEF) * 4u; }
__device__ __forceinline__ uint32_t ws_byte(int buf) { return (uint32_t)((2 + buf) * TILEF) * 4u; }

// Async transposing copy of one BMxBK f32 tile. GVS addressing:
//   mem = SGPR64(tile base) + VGPR32(lane byte offset); LDS dst from VGPR.
// Lane offsets are shared by the X and W streams (same row stride K).
__device__ __forceinline__ void issue_tile(const float* sbase,
                                           const uint32_t* voff,
                                           const uint32_t* ldsoff,
                                           uint32_t lds_base_bytes) {
#pragma unroll
  for (int i = 0; i < 8; ++i) {
    const uint32_t l = lds_base_bytes + ldsoff[i];
    asm volatile("global_load_async_to_lds_b32 %0, %1, %2"
                 :: "v"(l), "v"(voff[i]), "s"(sbase)
                 : "memory");
  }
}

template <int RED>
__device__ __forceinline__ float red3(float acc, float v0, float v1) {
  // Associative tropical reduction; min(acc,min(v0,v1)) -> v_min3_num_f32.
  return (RED == 0) ? __builtin_fminf(acc, __builtin_fminf(v0, v1))
                    : __builtin_fmaxf(acc, __builtin_fmaxf(v0, v1));
}

// RED: 0 = min-plus, 1 = max-plus.  COMBINE: 0 none, 1 min(aux), 2 max(aux).
template <int RED, int COMBINE, bool BIAS, int K, int N>
__global__ __launch_bounds__(256) void tropical_gemm(
    const float* __restrict__ X,    // [1024, K]
    const float* __restrict__ W,    // [N, K]
    const float* __restrict__ Bv,   // [N] or null
    const float* __restrict__ Aux,  // [1024, N] or null
    float* __restrict__ Out) {      // [1024, N]
  extern __shared__ float smem[];

  const int tid = threadIdx.x;
  const int tx = tid & 15;          // 16 threads across N
  const int ty = tid >> 4;          // 16 threads across M
  const int mBlk = blockIdx.y * BM;
  const int nBlk = blockIdx.x * BN;

  const float* xTile = X + (size_t)mBlk * K;
  const float* wTile = W + (size_t)nBlk * K;

  // Per-lane async-copy offsets (element e = tid + 256*i of the 64x32 tile):
  //   global: row r = e>>5 (K = power of two -> shift), col kk = e&31
  //   LDS   : transposed [kk][r] with padded stride.
  uint32_t voff[8], ldsoff[8];
  {
    const int r0 = tid >> 5, kk0 = tid & 31;
#pragma unroll
    for (int i = 0; i < 8; ++i) {
      const int r = r0 + 8 * i;
      voff[i]   = (uint32_t)(r * K + kk0) * 4u;
      ldsoff[i] = (uint32_t)(kk0 * LDST + r) * 4u;
    }
  }

  const float init = (RED == 0) ? __builtin_inff() : -__builtin_inff();
  float acc[4][4];
#pragma unroll
  for (int i = 0; i < 4; ++i)
#pragma unroll
    for (int j = 0; j < 4; ++j) acc[i][j] = init;

  constexpr int NT = K / BK;

  // Pipeline prologue: tile 0 into buffer 0.
  issue_tile(xTile, voff, ldsoff, xs_byte(0));
  issue_tile(wTile, voff, ldsoff, ws_byte(0));

#pragma unroll 2
  for (int t = 0; t < NT; ++t) {
    const int cur = t & 1;
    if (t + 1 < NT) {
      // Prefetch tile t+1 into the other buffer, then wait until only those
      // 16 freshly-issued ops remain outstanding (async loads complete in
      // order => tile t has fully landed in LDS).
      issue_tile(xTile + (size_t)(t + 1) * BK, voff, ldsoff, xs_byte(cur ^ 1));
      issue_tile(wTile + (size_t)(t + 1) * BK, voff, ldsoff, ws_byte(cur ^ 1));
      asm volatile("s_wait_asynccnt 16" ::: "memory");
    } else {
      asm volatile("s_wait_asynccnt 0" ::: "memory");
    }
    __syncthreads();  // tile t visible to all waves

    const float* xs = smem + (size_t)cur * TILEF + ty * 4;
    const float* ws = smem + (size_t)(2 + cur) * TILEF + tx * 4;
#pragma unroll
    for (int kk = 0; kk < BK; kk += 2) {
      const float4 a0 = *(const float4*)(xs + kk * LDST);        // ds_load_b128
      const float4 b0 = *(const float4*)(ws + kk * LDST);        // ds_load_b128
      const float4 a1 = *(const float4*)(xs + (kk + 1) * LDST);  // ds_load_b128
      const float4 b1 = *(const float4*)(ws + (kk + 1) * LDST);  // ds_load_b128
      const float av0[4] = {a0.x, a0.y, a0.z, a0.w};
      const float bv0[4] = {b0.x, b0.y, b0.z, b0.w};
      const float av1[4] = {a1.x, a1.y, a1.z, a1.w};
      const float bv1[4] = {b1.x, b1.y, b1.z, b1.w};
#pragma unroll
      for (int i = 0; i < 4; ++i)
#pragma unroll
        for (int j = 0; j < 4; ++j)
          acc[i][j] = red3<RED>(acc[i][j], av0[i] + bv0[j], av1[i] + bv1[j]);
    }
    __syncthreads();  // everyone done with buffer `cur` before t+2 overwrites it
  }

  // Fused epilogue: bias + shortcut combine + b128 stores.
  const int m0 = mBlk + ty * 4;
  const int n0 = nBlk + tx * 4;
  float4 bb = {0.f, 0.f, 0.f, 0.f};
  if (BIAS) bb = *(const float4*)(Bv + n0);
#pragma unroll
  for (int i = 0; i < 4; ++i) {
    float4 v;
    v.x = acc[i][0]; v.y = acc[i][1]; v.z = acc[i][2]; v.w = acc[i][3];
    if (BIAS) { v.x += bb.x; v.y += bb.y; v.z += bb.z; v.w += bb.w; }
    if (COMBINE != 0) {
      const float4 s = *(const float4*)(Aux + (size_t)(m0 + i) * N + n0);
      if (COMBINE == 1) {
        v.x = __builtin_fminf(v.x, s.x); v.y = __builtin_fminf(v.y, s.y);
        v.z = __builtin_fminf(v.z, s.z); v.w = __builtin_fminf(v.w, s.w);
      } else {
        v.x = __builtin_fmaxf(v.x, s.x); v.y = __builtin_fmaxf(v.y, s.y);
        v.z = __builtin_fmaxf(v.z, s.z); v.w = __builtin_fmaxf(v.w, s.w);
      }
    }
    *(float4*)(Out + (size_t)(m0 + i) * N + n0) = v;
  }
}

// ---------------------------------------------------------------------------
extern "C" void kernel_launch(void* const* d_in, const int* in_sizes, int n_in,
                              void* d_out, int out_size, void* d_ws, size_t ws_size,
                              hipStream_t stream) {
  const float* x      = (const float*)d_in[0];
  const float* l1_w1  = (const float*)d_in[1];
  const float* l1_w2  = (const float*)d_in[2];
  const float* l1_b2  = (const float*)d_in[3];
  const float* l2_w1  = (const float*)d_in[4];
  const float* l2_w2  = (const float*)d_in[5];
  const float* l2_b2  = (const float*)d_in[6];
  const float* l3_w1  = (const float*)d_in[7];
  const float* l3_w2  = (const float*)d_in[8];
  const float* l3_b2  = (const float*)d_in[9];
  const float* l4_w1  = (const float*)d_in[10];
  const float* l4_w2  = (const float*)d_in[11];
  const float* l4_b2  = (const float*)d_in[12];
  const float* l4_sw1 = (const float*)d_in[13];
  const float* l4_sw2 = (const float*)d_in[14];
  const float* l4_sb2 = (const float*)d_in[15];
  const float* sc_w1  = (const float*)d_in[16];
  const float* sc_w2  = (const float*)d_in[17];
  const float* sc_b2  = (const float*)d_in[18];
  float* out = (float*)d_out;

  // 4 ping-pong activation buffers, each sized for the widest layer (1024x512).
  const size_t BUF = 1024u * 512u;
  float* bufT = (float*)d_ws;
  float* bufA = bufT + BUF;
  float* bufB = bufA + BUF;
  float* bufC = bufB + BUF;

  const dim3 blk(256);
  const dim3 g256(256 / BN, 1024 / BM);  // (4, 16)
  const dim3 g512(512 / BN, 1024 / BM);  // (8, 16)

  // layer1: o1 = min( max_plus(min_plus(x, w1), w2) + b2 , x )
  tropical_gemm<0, 0, false, 256, 256><<<g256, blk, SMEM_BYTES, stream>>>(x,    l1_w1, nullptr, nullptr, bufT);
  tropical_gemm<1, 1, true,  256, 256><<<g256, blk, SMEM_BYTES, stream>>>(bufT, l1_w2, l1_b2,   x,       bufA);
  // layer2
  tropical_gemm<0, 0, false, 256, 256><<<g256, blk, SMEM_BYTES, stream>>>(bufA, l2_w1, nullptr, nullptr, bufT);
  tropical_gemm<1, 1, true,  256, 256><<<g256, blk, SMEM_BYTES, stream>>>(bufT, l2_w2, l2_b2,   bufA,    bufB);
  // layer3
  tropical_gemm<0, 0, false, 256, 256><<<g256, blk, SMEM_BYTES, stream>>>(bufB, l3_w1, nullptr, nullptr, bufT);
  tropical_gemm<1, 1, true,  256, 256><<<g256, blk, SMEM_BYTES, stream>>>(bufT, l3_w2, l3_b2,   bufB,    bufA); // o3
  // layer4 main path: main = max_plus(min_plus(o3, w1), w2) + b2
  tropical_gemm<0, 0, false, 256, 512><<<g512, blk, SMEM_BYTES, stream>>>(bufA, l4_w1, nullptr, nullptr, bufT);
  tropical_gemm<1, 0, true,  512, 512><<<g512, blk, SMEM_BYTES, stream>>>(bufT, l4_w2, l4_b2,   nullptr, bufB);
  // layer4 shortcut path, fused min with main: o4 = min(shortcut, main)
  tropical_gemm<0, 0, false, 256, 512><<<g512, blk, SMEM_BYTES, stream>>>(bufA, l4_sw1, nullptr, nullptr, bufT);
  tropical_gemm<1, 1, true,  512, 512><<<g512, blk, SMEM_BYTES, stream>>>(bufT, l4_sw2, l4_sb2, bufB,    bufC);
  // bottleneck shortcut from x, fused max with o4 -> final output
  tropical_gemm<0, 0, false, 256, 512><<<g512, blk, SMEM_BYTES, stream>>>(x,    sc_w1, nullptr, nullptr, bufT);
  tropical_gemm<1, 2, true,  512, 512><<<g512, blk, SMEM_BYTES, stream>>>(bufT, sc_w2, sc_b2,   bufC,    out);

  (void)in_sizes; (void)n_in; (void)out_size; (void)ws_size;
}